// DSC_11536282157800
// MI455X (gfx1250) — compile-verified
//
#include <hip/hip_runtime.h>

typedef __attribute__((ext_vector_type(2))) float v2f;
typedef __attribute__((ext_vector_type(8))) float v8f;

// Problem constants (from reference): D=512, N=P=256, H=16, M_LEN=24, L=2048
#define MATS (512l * 512l)

// ---------------------------------------------------------------------------
// Generic f32 GEMM via V_WMMA_F32_16X16X4_F32.
// C[M,N] (+)= A[M,K] @ B[K,N]
//   A addr: A + m*lda + k               (row-major, contiguous k; lda % 4 == 0)
//   B addr: B + k*ldb + n*bcs           (generalized: supports strided cols)
//   C addr: C + m*ldc + n
// Batched via grid.z with per-operand strides. K must be a multiple of 16.
// Block: 256 threads (8 waves), tile 128x64, K-step 16.
// LDS layouts give 8B-aligned v2f fragment loads (single ds_load_b64 each,
// no register-pair shuffles before v_wmma).
// ---------------------------------------------------------------------------
__global__ __launch_bounds__(256) void gemm_wmma_f32(
    const float* __restrict__ A, long lda, long strideA,
    const float* __restrict__ B, long ldb, long bcs, long strideB,
    float* __restrict__ C, long ldc, long strideC,
    int M, int N, int K, int accumulate)
{
  __shared__ float sA[128][18];   // even stride: aligned v2f, conflict-free
  __shared__ float sBT[64][18];   // B staged transposed: sBT[n][k]

  A += (long)blockIdx.z * strideA;
  B += (long)blockIdx.z * strideB;
  C += (long)blockIdx.z * strideC;

  const int m0   = blockIdx.y * 128;
  const int n0   = blockIdx.x * 64;
  const int tid  = threadIdx.x;
  const int wave = tid >> 5;
  const int lane = tid & 31;
  const int half = lane >> 4;   // 0: lanes 0-15, 1: lanes 16-31
  const int l16  = lane & 15;

  v8f acc[4] = {};              // four 16x16 f32 tiles along N per wave

  for (int k0 = 0; k0 < K; k0 += 16) {
    // Stage A panel 128x16 as float4 global loads (2 per thread)
#pragma unroll
    for (int r = 0; r < 2; ++r) {
      int idx4 = tid + (r << 8);          // 0..511
      int row  = idx4 >> 2;               // 0..127
      int c4   = (idx4 & 3) << 2;         // 0,4,8,12
      int gm   = m0 + row;
      float4 v = make_float4(0.f, 0.f, 0.f, 0.f);
      if (gm < M) v = *(const float4*)(A + (long)gm * lda + (k0 + c4));
      v2f lo; lo.x = v.x; lo.y = v.y;
      v2f hi; hi.x = v.z; hi.y = v.w;
      *(v2f*)&sA[row][c4]     = lo;
      *(v2f*)&sA[row][c4 + 2] = hi;
    }
    // Stage B panel 16x64 transposed (4 scalars per thread, generic bcs)
#pragma unroll
    for (int r = 0; r < 4; ++r) {
      int row = (tid >> 6) + (r << 2);    // k within panel: 0..15
      int col = tid & 63;                 // n within panel
      int gn  = n0 + col;
      float v = 0.f;
      if (gn < N) v = B[(long)(k0 + row) * ldb + (long)gn * bcs];
      sBT[col][row] = v;
    }
    __syncthreads();

#pragma unroll
    for (int kk = 0; kk < 16; kk += 4) {
      // A fragment 16x4: lane holds A[m=l16][kk+2*half .. +1]  (one b64 load)
      v2f a = *(const v2f*)&sA[(wave << 4) + l16][kk + (half << 1)];
#pragma unroll
      for (int nt = 0; nt < 4; ++nt) {
        // B fragment 4x16: lane holds B[kk+2*half .. +1][n=l16] (one b64 load)
        v2f b = *(const v2f*)&sBT[(nt << 4) + l16][kk + (half << 1)];
        acc[nt] = __builtin_amdgcn_wmma_f32_16x16x4_f32(
            false, a, false, b, (short)0, acc[nt], false, false);
      }
    }
    __syncthreads();
  }

  // Store: C/D layout -> VGPR v holds row (v + 8*half), col l16
#pragma unroll
  for (int nt = 0; nt < 4; ++nt) {
#pragma unroll
    for (int v = 0; v < 8; ++v) {
      int gm = m0 + (wave << 4) + v + (half << 3);
      int gn = n0 + (nt << 4) + l16;
      if (gm < M && gn < N) {
        long idx  = (long)gm * ldc + gn;
        float val = acc[nt][v];
        C[idx] = accumulate ? (C[idx] + val) : val;
      }
    }
  }
}

// ---------------------------------------------------------------------------
// Small helper kernels
// ---------------------------------------------------------------------------
__global__ void ident512(float* __restrict__ d) {
  d[blockIdx.x * 512 + threadIdx.x] = (blockIdx.x == (int)threadIdx.x) ? 1.f : 0.f;
}

__global__ void transposeU(const float* __restrict__ u, float* __restrict__ ut) {
  int p = blockIdx.x;                       // 256 rows of U^T
  for (int t = threadIdx.x; t < 2048; t += 256)
    ut[p * 2048 + t] = u[t * 256 + p];      // coalesced writes
}

__global__ void coeffs_k(const float* __restrict__ lambda_e, const float* __restrict__ sigma,
                         float* __restrict__ lam4, float* __restrict__ sig4) {
  int t = threadIdx.x;
  if (t < 16) lam4[t] = powf(lambda_e[t], 0.25f);
  else if (t < 33) sig4[t - 16] = powf(sigma[t - 16], 0.25f);
}

// y_nat[p] = y_hist[L-1,p] - C[p,:] . s
__global__ void ynat_k(const float* __restrict__ yh, const float* __restrict__ Cm,
                       const float* __restrict__ s, float* __restrict__ ynat) {
  int p = threadIdx.x;
  float acc = 0.f;
  for (int d = 0; d < 512; ++d) acc += Cm[p * 512 + d] * s[d];
  ynat[p] = yh[2047 * 256 + p] - acc;
}

// z_i[p] = sum_j phi_t[j,i] * Yrev[1+j,p];  w_l[p] = sum_k phi[k,l] * Yrev[k,p]
__global__ void zw_k(const float* __restrict__ ynh, const float* __restrict__ phi_t,
                     const float* __restrict__ phi, float* __restrict__ zv,
                     float* __restrict__ wv) {
  int b = blockIdx.x, p = threadIdx.x;
  if (b < 16) {
    float s = 0.f;
    for (int j = 0; j < 24; ++j) s += phi_t[j * 16 + b] * ynh[(2046 - j) * 256 + p];
    zv[b * 256 + p] = s;
  } else {
    int l = b - 16;
    float s = 0.f;
    for (int k = 0; k < 25; ++k) s += phi[k * 17 + l] * ynh[(2047 - k) * 256 + p];
    wv[l * 256 + p] = s;
  }
}

// Q[i,k,p] = sum_j phi_t[j,i] * Yjk[j,k,p],  Yjk[j,k] = ynh[L-3-j-k]
__global__ void q_k(const float* __restrict__ ynh, const float* __restrict__ phi_t,
                    float* __restrict__ Q) {
  int b = blockIdx.x, p = threadIdx.x;    // b = i*25 + k
  int i = b / 25, k = b % 25;
  float s = 0.f;
  for (int j = 0; j < 24; ++j) s += phi_t[j * 16 + i] * ynh[(2045 - j - k) * 256 + p];
  Q[b * 256 + p] = s;
}

// Zv[i,l,p] = sum_k phi[k,l] * Q[i,k,p]
__global__ void zv_k(const float* __restrict__ phi, const float* __restrict__ Q,
                     float* __restrict__ Zv) {
  int b = blockIdx.x, p = threadIdx.x;    // b = i*17 + l
  int i = b / 17, l = b % 17;
  float s = 0.f;
  for (int k = 0; k < 25; ++k) s += phi[k * 17 + l] * Q[(i * 25 + k) * 256 + p];
  Zv[b * 256 + p] = s;
}

// 322 independent 256x256 matvecs (all M/M_bar/G terms), wave-per-row,
// coalesced mat reads, deterministic (no atomics). upart[b,n] written.
__global__ __launch_bounds__(256) void batched_matvec(
    const float* __restrict__ Mt, const float* __restrict__ Mb,
    const float* __restrict__ G, const float* __restrict__ ynh,
    const float* __restrict__ uh, const float* __restrict__ zv,
    const float* __restrict__ wv, const float* __restrict__ Zv,
    const float* __restrict__ lam4, const float* __restrict__ sig4,
    float* __restrict__ upart)
{
  __shared__ float sv[256];
  const int b = blockIdx.x;
  const float* mat;
  const float* vec;
  float coeff;
  if (b == 0)        { mat = Mb;                          vec = ynh + 2047 * 256;        coeff = 1.f; }
  else if (b <= 16)  { int i = b - 1;
                       mat = Mb + (long)b * 65536;        vec = zv + i * 256;            coeff = lam4[i]; }
  else if (b <= 33)  { int l = b - 17;
                       mat = Mt + (long)l * 65536;        vec = wv + l * 256;            coeff = sig4[l]; }
  else if (b <= 305) { int idx = b - 34, i = idx / 17, l = idx % 17;
                       mat = Mt + (long)((1 + i) * 17 + l) * 65536;
                       vec = Zv + idx * 256;              coeff = lam4[i] * sig4[l]; }
  else               { int k = b - 306;
                       mat = G + (long)k * 65536;         vec = uh + (2047 - k) * 256;   coeff = 1.f; }

  const int tid = threadIdx.x;
  sv[tid] = vec[tid];
  __syncthreads();

  const int wave = tid >> 5, lane = tid & 31;
  for (int row = wave; row < 256; row += 8) {
    const float* mrow = mat + (long)row * 256;
    if (row + 8 < 256)
      __builtin_prefetch(mat + (long)(row + 8) * 256 + (lane << 3), 0, 0);
    float s = 0.f;
#pragma unroll
    for (int q = 0; q < 8; ++q) s += mrow[lane + 32 * q] * sv[lane + 32 * q];
#pragma unroll
    for (int off = 16; off > 0; off >>= 1) s += __shfl_xor(s, off, 32);
    if (lane == 0) upart[(long)b * 256 + row] = coeff * s;
  }
}

// out = [y_nat | pred | u_t], fixed-order reductions (deterministic)
__global__ void assemble_k(const float* __restrict__ ynat, const float* __restrict__ upart,
                           float* __restrict__ out) {
  int p = threadIdx.x;
  out[p] = ynat[p];
  float pr = ynat[p];
  for (int b = 306; b < 322; ++b) pr += upart[(long)b * 256 + p];
  out[256 + p] = pr;
  float ut = 0.f;
  for (int b = 0; b < 306; ++b) ut += upart[(long)b * 256 + p];
  out[512 + p] = ut;
}

// ---------------------------------------------------------------------------
static inline void launch_gemm(hipStream_t s,
    const float* A, long lda, long sA,
    const float* B, long ldb, long bcs, long sB,
    float* C, long ldc, long sC,
    int M, int N, int K, int acc, int batch) {
  dim3 grid((N + 63) / 64, (M + 127) / 128, batch);
  gemm_wmma_f32<<<grid, 256, 0, s>>>(A, lda, sA, B, ldb, bcs, sB, C, ldc, sC, M, N, K, acc);
}

extern "C" void kernel_launch(void* const* d_in, const int* in_sizes, int n_in,
                              void* d_out, int out_size, void* d_ws, size_t ws_size,
                              hipStream_t stream) {
  const float* A        = (const float*)d_in[0];   // (512,512)
  const float* B        = (const float*)d_in[1];   // (512,256)
  const float* Cm       = (const float*)d_in[2];   // (256,512)
  const float* Mt       = (const float*)d_in[3];   // (17,17,256,256)
  const float* Mb       = (const float*)d_in[4];   // (17,256,256)
  const float* sigma    = (const float*)d_in[5];   // (17,)
  const float* phi      = (const float*)d_in[6];   // (25,17)
  const float* lambda_e = (const float*)d_in[7];   // (16,)
  const float* phi_t    = (const float*)d_in[8];   // (24,16)
  const float* yh       = (const float*)d_in[9];   // (2048,256)
  const float* uh       = (const float*)d_in[10];  // (2048,256)
  const float* ynh      = (const float*)d_in[11];  // (2048,256)
  float* out = (float*)d_out;

  // Workspace layout (floats)
  float* ws    = (float*)d_ws;
  float* Apow  = ws;                      // A^0..A^16            17*MATS
  float* A16   = Apow + 17 * MATS;        // (A^16)^0..^15        16*MATS
  float* A256  = A16 + 16 * MATS;         // (A^256)^0..^7         8*MATS
  float* UT    = A256 + 8 * MATS;         // U^T (256x2048)       524288
  float* Z     = UT + 524288;             // B@U^T (512x2048)    1048576
  float* V     = Z + 1048576;             // chunk sums (512x128)  65536
  float* V2    = V + 65536;               // (512x8)                4096
  float* svec  = V2 + 4096;               // final state s (512)     512
  float* CA    = svec + 512;              // C@A^k stack (16x256x512)
  float* G     = CA + 2097152;            // G stack (16x256x256)
  float* ynat  = G + 1048576;             // (256)
  float* lam4  = ynat + 256;              // (16, padded)
  float* sig4  = lam4 + 32;               // (17, padded)
  float* zv    = sig4 + 32;               // (16x256)
  float* wv    = zv + 4096;               // (17x256)
  float* Q     = wv + 4352;               // (16x25x256)
  float* Zv    = Q + 102400;              // (16x17x256)
  float* upart = Zv + 69632;              // (322x256)

  // --- coefficients + identity matrices ---
  coeffs_k<<<1, 64, 0, stream>>>(lambda_e, sigma, lam4, sig4);
  ident512<<<512, 512, 0, stream>>>(Apow);
  ident512<<<512, 512, 0, stream>>>(A16);
  ident512<<<512, 512, 0, stream>>>(A256);

  // --- A powers A^1..A^16 (also used for G) ---
  hipMemcpyAsync(Apow + MATS, A, MATS * sizeof(float), hipMemcpyDeviceToDevice, stream);
  for (int k = 2; k <= 16; ++k)
    launch_gemm(stream, Apow + (k - 1) * MATS, 512, 0, A, 512, 1, 0,
                Apow + (long)k * MATS, 512, 0, 512, 512, 512, 0, 1);

  // --- Z = B @ U^T : z_t = B u_t for all t as one GEMM ---
  transposeU<<<256, 256, 0, stream>>>(uh, UT);
  launch_gemm(stream, B, 256, 0, UT, 2048, 1, 0, Z, 2048, 0, 512, 2048, 256, 0, 1);

  // --- level 1: V[:,j] = sum_i A^(15-i) @ Z[:,16j+i]  (stream-serialized accum) ---
  for (int i = 0; i < 16; ++i)
    launch_gemm(stream, Apow + (long)(15 - i) * MATS, 512, 0,
                Z + i, 2048, 16, 0, V, 128, 0, 512, 128, 512, i > 0, 1);

  // --- powers of A^16 ---
  hipMemcpyAsync(A16 + MATS, Apow + 16 * MATS, MATS * sizeof(float),
                 hipMemcpyDeviceToDevice, stream);
  for (int k = 2; k <= 15; ++k)
    launch_gemm(stream, A16 + (k - 1) * MATS, 512, 0, Apow + 16 * MATS, 512, 1, 0,
                A16 + (long)k * MATS, 512, 0, 512, 512, 512, 0, 1);

  // --- level 2: V2[:,m] = sum_i (A^16)^(15-i) @ V[:,16m+i] ---
  for (int i = 0; i < 16; ++i)
    launch_gemm(stream, A16 + (long)(15 - i) * MATS, 512, 0,
                V + i, 128, 16, 0, V2, 8, 0, 512, 8, 512, i > 0, 1);

  // --- powers of A^256 ---
  launch_gemm(stream, A16 + 15 * MATS, 512, 0, Apow + 16 * MATS, 512, 1, 0,
              A256 + MATS, 512, 0, 512, 512, 512, 0, 1);      // A^256
  for (int k = 2; k <= 7; ++k)
    launch_gemm(stream, A256 + (k - 1) * MATS, 512, 0, A256 + MATS, 512, 1, 0,
                A256 + (long)k * MATS, 512, 0, 512, 512, 512, 0, 1);

  // --- level 3: s = sum_m (A^256)^(7-m) @ V2[:,m] ---
  for (int m = 0; m < 8; ++m)
    launch_gemm(stream, A256 + (long)(7 - m) * MATS, 512, 0,
                V2 + m, 8, 1, 0, svec, 1, 0, 512, 1, 512, m > 0, 1);

  // --- y_nat ---
  ynat_k<<<1, 256, 0, stream>>>(yh, Cm, svec, ynat);

  // --- G_k = (C @ A^k) @ B, k=0..15, batched GEMMs ---
  launch_gemm(stream, Cm, 512, 0, Apow, 512, 1, MATS, CA, 512, 131072,
              256, 512, 512, 0, 16);
  launch_gemm(stream, CA, 512, 131072, B, 256, 1, 0, G, 256, 65536,
              256, 256, 512, 0, 16);

  // --- small Y-history projections ---
  zw_k<<<33, 256, 0, stream>>>(ynh, phi_t, phi, zv, wv);
  q_k<<<400, 256, 0, stream>>>(ynh, phi_t, Q);
  zv_k<<<272, 256, 0, stream>>>(phi, Q, Zv);

  // --- all 322 matvec terms (streams M once, memory-bound) ---
  batched_matvec<<<322, 256, 0, stream>>>(Mt, Mb, G, ynh, uh, zv, wv, Zv,
                                          lam4, sig4, upart);

  // --- final deterministic reductions into out[768] ---
  assemble_k<<<1, 256, 0, stream>>>(ynat, upart, out);

  (void)in_sizes; (void)n_in; (void)out_size; (void)ws_size;
}